// OnnxNonMaxSuppression_72722386255884
// MI455X (gfx1250) — compile-verified
//
#include <hip/hip_runtime.h>
#include <stdint.h>

// ---------------------------------------------------------------------------
// ONNX NonMaxSuppression  (B=8, C=20, N=2000, iou_thr=0.5, score_thr=0.1)
// One workgroup per (batch, class). Boxes staged to LDS via CDNA5
// global_load_async_to_lds_b128, bitonic stable sort of score keys,
// then greedy suppression with IoU recomputed from LDS.
// ---------------------------------------------------------------------------

#define Bn 8
#define Cn 20
#define Nn 2000
#define NP 2048          // next pow2 for bitonic sort
#define NTHREADS 256
#define IOU_THR 0.5f
#define SCORE_THR 0.1f

#if __has_builtin(__builtin_amdgcn_global_load_async_to_lds_b128)
#define HAVE_ASYNC_LDS 1
#else
#define HAVE_ASYNC_LDS 0
#endif

typedef int v4i __attribute__((ext_vector_type(4)));

// Monotone map float -> uint32 (ascending float order == ascending uint order)
__device__ __forceinline__ uint32_t ordered_key(float f) {
    uint32_t u = __float_as_uint(f);
    return (u & 0x80000000u) ? ~u : (u ^ 0x80000000u);
}

__global__ __launch_bounds__(NTHREADS)
void nms_kernel(const float* __restrict__ boxes,   // [B,N,4] (y1,x1,y2,x2)
                const float* __restrict__ scores,  // [B,C,N]
                int* __restrict__ out)             // [B,C,N,3] int32
{
    __shared__ __align__(16) float rawbox[Nn * 4];  // 32000 B (y1,x1,y2,x2)
    __shared__ float    area[Nn];                   //  8000 B
    __shared__ uint64_t keys[NP];                   // 16384 B
    __shared__ uint16_t ord[NP];                    //  4096 B
    __shared__ uint8_t  keep[NP];                   //  2048 B

    const int bc  = blockIdx.x;
    const int b   = bc / Cn;
    const int c   = bc % Cn;
    const int tid = threadIdx.x;

    const float* bb = boxes  + (size_t)b * Nn * 4;
    const float* ss = scores + ((size_t)b * Cn + c) * Nn;

    // Hint the score vector into cache early (global_prefetch_b8 on gfx1250).
    __builtin_prefetch(ss, 0, 1);
    __builtin_prefetch(ss + 1024, 0, 1);

    // ---- Stage boxes [N,4] into LDS ------------------------------------
#if HAVE_ASYNC_LDS
    {
        // 32000 bytes = 2000 x b128 chunks: async DMA global -> LDS
        auto gsrc = (__attribute__((address_space(1))) v4i*)(void*)const_cast<float*>(bb);
        auto ldst = (__attribute__((address_space(3))) v4i*)(void*)&rawbox[0];
        const int nchunks = (Nn * 4 * 4) / 16;  // 2000
        for (int n = tid; n < nchunks; n += NTHREADS) {
            __builtin_amdgcn_global_load_async_to_lds_b128(gsrc + n, ldst + n, 0, 0);
        }
#if __has_builtin(__builtin_amdgcn_s_wait_asynccnt)
        __builtin_amdgcn_s_wait_asynccnt(0);
#else
        asm volatile("s_wait_asynccnt 0" ::: "memory");
#endif
    }
#else
    {
        const float4* src = (const float4*)bb;
        float4*       dst = (float4*)&rawbox[0];
        for (int n = tid; n < Nn; n += NTHREADS) dst[n] = src[n];  // b128 copy
    }
#endif
    __syncthreads();

    // Areas (boxes are min/max-constructed, so x2>=x1, y2>=y1)
    for (int n = tid; n < Nn; n += NTHREADS) {
        float y1 = rawbox[n * 4 + 0], x1 = rawbox[n * 4 + 1];
        float y2 = rawbox[n * 4 + 2], x2 = rawbox[n * 4 + 3];
        area[n] = (x2 - x1) * (y2 - y1);
    }

    // ---- Build sort keys: (ordered(-score or +inf) << 32) | index -------
    // Reproduces jnp.argsort(where(valid, -score, inf)) with stable ties.
    for (int n = tid; n < NP; n += NTHREADS) {
        uint64_t k;
        if (n < Nn) {
            float s = ss[n];
            float f = (s > SCORE_THR) ? -s : __builtin_inff();
            k = ((uint64_t)ordered_key(f) << 32) | (uint32_t)n;
        } else {
            k = ~0ull;  // padding sorts strictly last
        }
        keys[n] = k;
    }

    // ---- Bitonic sort (ascending) of 2048 uint64 keys -------------------
    for (unsigned kk = 2; kk <= NP; kk <<= 1) {
        for (unsigned j = kk >> 1; j > 0; j >>= 1) {
            __syncthreads();
            for (unsigned i = tid; i < NP; i += NTHREADS) {
                unsigned ix = i ^ j;
                if (ix > i) {
                    uint64_t a  = keys[i];
                    uint64_t bv = keys[ix];
                    bool asc = ((i & kk) == 0);
                    if ((a > bv) == asc) { keys[i] = bv; keys[ix] = a; }
                }
            }
        }
    }
    __syncthreads();

    // ---- Extract order + initial keep mask ------------------------------
    // valid entries have key-hi < ordered(+inf)=0xFF800000 (padding = 0xFFFFFFFF)
    for (int n = tid; n < NP; n += NTHREADS) {
        uint64_t k = keys[n];
        ord[n]  = (uint16_t)(k & 0xFFFFu);
        keep[n] = ((uint32_t)(k >> 32) < 0xFF800000u) ? 1 : 0;
    }

    // ---- Greedy suppression --------------------------------------------
    for (int i = 0; i < Nn; ++i) {
        __syncthreads();                 // keep[i] finalized by iters < i
        if (!keep[i]) continue;          // uniform across block (LDS value)
        const int oi = ord[i];
        const float y1i = rawbox[oi * 4 + 0], x1i = rawbox[oi * 4 + 1];
        const float y2i = rawbox[oi * 4 + 2], x2i = rawbox[oi * 4 + 3];
        const float ai  = area[oi];
        for (int p = i + 1 + tid; p < Nn; p += NTHREADS) {
            if (!keep[p]) continue;
            const int oj = ord[p];
            float xx1 = fmaxf(x1i, rawbox[oj * 4 + 1]);
            float yy1 = fmaxf(y1i, rawbox[oj * 4 + 0]);
            float xx2 = fminf(x2i, rawbox[oj * 4 + 3]);
            float yy2 = fminf(y2i, rawbox[oj * 4 + 2]);
            float w = fmaxf(xx2 - xx1, 0.0f);
            float h = fmaxf(yy2 - yy1, 0.0f);
            float inter = w * h;
            float uni   = ai + area[oj] - inter;
            float iou   = inter / fmaxf(uni, 1e-9f);
            if (iou > IOU_THR) keep[p] = 0;
        }
    }
    __syncthreads();

    // ---- Emit padded triples -------------------------------------------
    int* o = out + ((size_t)bc) * Nn * 3;
    for (int p = tid; p < Nn; p += NTHREADS) {
        int v0 = -1, v1 = -1, v2 = -1;
        if (keep[p]) { v0 = b; v1 = c; v2 = (int)ord[p]; }
        o[p * 3 + 0] = v0;
        o[p * 3 + 1] = v1;
        o[p * 3 + 2] = v2;
    }
}

extern "C" void kernel_launch(void* const* d_in, const int* in_sizes, int n_in,
                              void* d_out, int out_size, void* d_ws, size_t ws_size,
                              hipStream_t stream) {
    (void)in_sizes; (void)n_in; (void)out_size; (void)d_ws; (void)ws_size;
    const float* boxes  = (const float*)d_in[0];  // [B,N,4] float32
    const float* scores = (const float*)d_in[1];  // [B,C,N] float32
    int* out = (int*)d_out;                       // [B,C,N,3] int32

    dim3 grid(Bn * Cn);
    dim3 block(NTHREADS);
    nms_kernel<<<grid, block, 0, stream>>>(boxes, scores, out);
}